// DG_29119878267409
// MI455X (gfx1250) — compile-verified
//
#include <hip/hip_runtime.h>

typedef float v2f __attribute__((ext_vector_type(2)));
typedef float v8f __attribute__((ext_vector_type(8)));

#define N_ROWS 8192
#define D_IN   1024
#define D_OUT  4096
#define GAMMA  0.01618f
#define SLOPE  0.01f
#define KSEL   10

// ---------------------------------------------------------------------------
// Kernel 1: zero the output (binary mask is mostly zeros; scatter ones later)
// ---------------------------------------------------------------------------
__global__ void DG_zero_kernel(float4* __restrict__ out, int n4) {
    int i = blockIdx.x * blockDim.x + threadIdx.x;
    int stride = gridDim.x * blockDim.x;
    for (; i < n4; i += stride) out[i] = make_float4(0.f, 0.f, 0.f, 0.f);
}

// ---------------------------------------------------------------------------
// Kernel 2: H = leaky_relu(X @ W^T + b), f32 WMMA 16x16x4.
// Each wave computes a 16(M) x 64(N) tile: one A operand reused across 4 WMMAs.
// A layout (16x4 f32): lane L -> M = L%16, VGPR j holds K = 2*(L/16)+j.
// B layout (4x16 f32): lane L -> N = L%16, VGPR j holds K = 2*(L/16)+j.
//   B[k][n] = W[n][k], so B loads have the same index structure as A but on W.
// C/D layout (16x16 f32): lane L, VGPR v -> M = v + 8*(L/16), N = L%16.
// ---------------------------------------------------------------------------
__global__ __launch_bounds__(256) void DG_fc_wmma_kernel(
    const float* __restrict__ X, const float* __restrict__ W,
    const float* __restrict__ bias, float* __restrict__ H)
{
    const int tid  = threadIdx.x;
    const int wave = tid >> 5;
    const int lane = tid & 31;
    const int half = lane >> 4;   // 0 or 1
    const int r16  = lane & 15;

    const int mBase = blockIdx.y * 16;                 // 512 M-tiles
    const int nBase = (blockIdx.x * 8 + wave) * 64;    // 8 waves * 64 cols = 512 cols/block

    const float* xrow = X + (size_t)(mBase + r16) * D_IN + 2 * half;
    const float* wrow = W + (size_t)(nBase + r16) * D_IN + 2 * half;

    v8f c0 = {}, c1 = {}, c2 = {}, c3 = {};

#pragma unroll 4
    for (int k0 = 0; k0 < D_IN; k0 += 4) {
        v2f a  = *(const v2f*)(xrow + k0);
        v2f b0 = *(const v2f*)(wrow + k0);
        v2f b1 = *(const v2f*)(wrow + 16 * D_IN + k0);
        v2f b2 = *(const v2f*)(wrow + 32 * D_IN + k0);
        v2f b3 = *(const v2f*)(wrow + 48 * D_IN + k0);
        c0 = __builtin_amdgcn_wmma_f32_16x16x4_f32(false, a, false, b0, (short)0, c0, false, false);
        c1 = __builtin_amdgcn_wmma_f32_16x16x4_f32(false, a, false, b1, (short)0, c1, false, false);
        c2 = __builtin_amdgcn_wmma_f32_16x16x4_f32(false, a, false, b2, (short)0, c2, false, false);
        c3 = __builtin_amdgcn_wmma_f32_16x16x4_f32(false, a, false, b3, (short)0, c3, false, false);
    }

    // Epilogue: bias + leaky_relu, store rows of H.
#pragma unroll
    for (int v = 0; v < 8; ++v) {
        const int m = mBase + v + 8 * half;
        float* hrow = H + (size_t)m * D_OUT + nBase + r16;
        float y0 = c0[v] + bias[nBase + r16];
        float y1 = c1[v] + bias[nBase + 16 + r16];
        float y2 = c2[v] + bias[nBase + 32 + r16];
        float y3 = c3[v] + bias[nBase + 48 + r16];
        hrow[0]  = y0 > 0.f ? y0 : SLOPE * y0;
        hrow[16] = y1 > 0.f ? y1 : SLOPE * y1;
        hrow[32] = y2 > 0.f ? y2 : SLOPE * y2;
        hrow[48] = y3 > 0.f ? y3 : SLOPE * y3;
    }
}

// ---------------------------------------------------------------------------
// Kernel 3: serial phi-recurrence scan. One workgroup, 1024 threads = 32 waves.
// Thread t owns columns 4t..4t+3 (phi kept in registers).
// Rows are double-buffered into LDS with GLOBAL_LOAD_ASYNC_TO_LDS_B128:
// each thread async-copies its 16B of row r+1 while row r is processed,
// then s_wait_asynccnt 0x1 guarantees row r's transfer completed (in-order).
// Per row: per-wave top-10 via shfl_xor argmax (no barriers inside a wave),
// then wave 0 merges 32x10 candidates -> global top-10. 3 barriers per row.
// ---------------------------------------------------------------------------
__global__ __launch_bounds__(1024) void DG_scan_topk_kernel(
    const float* __restrict__ H, float* __restrict__ out)
{
    __shared__ float rowbuf[2][D_OUT];   // 32 KB ping-pong row buffer
    __shared__ float wv[32 * KSEL];
    __shared__ int   wi[32 * KSEL];
    __shared__ float tv[KSEL];
    __shared__ int   ti[KSEL];

    const int t    = threadIdx.x;
    const int lane = t & 31;
    const int wave = t >> 5;
    const float NEG = -__builtin_inff();

    // LDS byte offsets of this thread's 16B slot in each buffer
    // (generic shared pointer keeps the LDS offset in its low 32 bits)
    const unsigned lds0 = (unsigned)(uintptr_t)(&rowbuf[0][4 * t]);
    const unsigned lds1 = (unsigned)(uintptr_t)(&rowbuf[1][4 * t]);

    float p0 = 1.f, p1 = 1.f, p2 = 1.f, p3 = 1.f;   // phi for columns 4t..4t+3

    // Prologue: async-load row 0 into buffer 0
    {
        const float* g0 = H + (size_t)4 * t;
        asm volatile("global_load_async_to_lds_b128 %0, %1, off"
                     :: "v"(lds0), "v"(g0) : "memory");
    }

    for (int r = 0; r < N_ROWS; ++r) {
        const int b = r & 1;
        // Issue async load of the next row into the other buffer.
        // (wraps to row 0 on the last iteration: harmless, keeps waits uniform)
        {
            const int rn = (r + 1) & (N_ROWS - 1);
            const float* gn = H + (size_t)rn * D_OUT + 4 * t;
            asm volatile("global_load_async_to_lds_b128 %0, %1, off"
                         :: "v"(b ? lds0 : lds1), "v"(gn) : "memory");
        }
        // Row r's transfer (older of the two outstanding) is now complete:
        asm volatile("s_wait_asynccnt 0x1" ::: "memory");

        const float4 h4 = *(const float4*)(&rowbuf[b][4 * t]);

        float m0 = h4.x * p0, m1 = h4.y * p1, m2 = h4.z * p2, m3 = h4.w * p3;

        // --- per-wave top-10 extraction (barrier-free, wave32 shuffles) ---
        for (int j = 0; j < KSEL; ++j) {
            float bv = m0; int bi = 4 * t;
            if (m1 > bv) { bv = m1; bi = 4 * t + 1; }
            if (m2 > bv) { bv = m2; bi = 4 * t + 2; }
            if (m3 > bv) { bv = m3; bi = 4 * t + 3; }
#pragma unroll
            for (int off = 16; off > 0; off >>= 1) {
                float ov = __shfl_xor(bv, off, 32);
                int   oi = __shfl_xor(bi, off, 32);
                if (ov > bv || (ov == bv && oi < bi)) { bv = ov; bi = oi; }
            }
            if (lane == 0) { wv[wave * KSEL + j] = bv; wi[wave * KSEL + j] = bi; }
            if ((bi >> 2) == t) {          // owner masks the extracted element
                const int c = bi & 3;
                if (c == 0) m0 = NEG; else if (c == 1) m1 = NEG;
                else if (c == 2) m2 = NEG; else m3 = NEG;
            }
        }
        __syncthreads();

        // --- wave 0 merges 32 sorted candidate lists into global top-10 ---
        if (wave == 0) {
            int p = 0;
            float hv = wv[lane * KSEL];
            int   hi = wi[lane * KSEL];
            for (int j = 0; j < KSEL; ++j) {
                float bv = hv; int bi = hi; int bl = lane;
#pragma unroll
                for (int off = 16; off > 0; off >>= 1) {
                    float ov = __shfl_xor(bv, off, 32);
                    int   oi = __shfl_xor(bi, off, 32);
                    int   ol = __shfl_xor(bl, off, 32);
                    if (ov > bv || (ov == bv && oi < bi)) { bv = ov; bi = oi; bl = ol; }
                }
                if (lane == 0) { tv[j] = bv; ti[j] = bi; }
                if (lane == bl) {            // winning list advances its head
                    ++p;
                    hv = (p < KSEL) ? wv[lane * KSEL + p] : NEG;
                    hi = (p < KSEL) ? wi[lane * KSEL + p] : 0x7fffffff;
                }
            }
        }
        __syncthreads();

        // --- phi recovery: phi = min(phi<1 ? phi+gamma : phi, 1) ---
        p0 = fminf(p0 < 1.f ? p0 + GAMMA : p0, 1.f);
        p1 = fminf(p1 < 1.f ? p1 + GAMMA : p1, 1.f);
        p2 = fminf(p2 < 1.f ? p2 + GAMMA : p2, 1.f);
        p3 = fminf(p3 < 1.f ? p3 + GAMMA : p3, 1.f);

        // --- scatter ones for top-10 entries with s>0, reset their phi ---
#pragma unroll
        for (int j = 0; j < KSEL; ++j) {
            const float v = tv[j];
            const int idx = ti[j];
            if (v > 0.f && (idx >> 2) == t) {
                out[(size_t)r * D_OUT + idx] = 1.0f;
                const int c = idx & 3;
                if (c == 0) p0 = 0.f; else if (c == 1) p1 = 0.f;
                else if (c == 2) p2 = 0.f; else p3 = 0.f;
            }
        }
        __syncthreads();   // protect tv/ti & wv/wi before next row overwrites
    }
}

// ---------------------------------------------------------------------------
extern "C" void kernel_launch(void* const* d_in, const int* in_sizes, int n_in,
                              void* d_out, int out_size, void* d_ws, size_t ws_size,
                              hipStream_t stream) {
    (void)in_sizes; (void)n_in; (void)out_size; (void)ws_size;
    const float* X    = (const float*)d_in[0];   // [8192,1024]
    const float* W    = (const float*)d_in[1];   // [4096,1024]
    const float* bias = (const float*)d_in[2];   // [4096]
    float* out = (float*)d_out;                  // [8192,4096] binary mask (f32)
    float* H   = (float*)d_ws;                   // [8192,4096] f32 intermediate (128 MiB)

    // 1) zero output at full device bandwidth
    DG_zero_kernel<<<4096, 256, 0, stream>>>((float4*)out, (N_ROWS * D_OUT) / 4);

    // 2) WMMA GEMM + bias + leaky_relu  (grid: 8 col-groups x 512 row-tiles)
    dim3 g(D_OUT / (64 * 8), N_ROWS / 16);
    DG_fc_wmma_kernel<<<g, 256, 0, stream>>>(X, W, bias, H);

    // 3) serial scan (single workgroup; recurrence over rows)
    DG_scan_topk_kernel<<<1, 1024, 0, stream>>>(H, out);
}